// RRNNforGRUCell_18004502905363
// MI455X (gfx1250) — compile-verified
//
#include <hip/hip_runtime.h>

// ---------------------------------------------------------------------------
// MI455X (gfx1250) RRNN-for-GRU cell.
//
// Roofline: each round streams 512MB of f32 weights for ~268 MFLOP -> ~100x
// memory bound.  Strategy: full-bandwidth GEMV streaming with b128 loads +
// LDS-staged s-vector + global prefetch; exact algebraic dedup of rounds
// (G0==G1==G2, G7==G3, r5 L-side is zero, x@L and G3@R reused) cuts the 18
// naive 256MB matrix-batch reads to 8 (~2.0GB ~= 88us at 23.3TB/s).
// The scoring MLP (4x4096 @ 4096x256) runs on v_wmma_f32_16x16x4_f32 with a
// zero-padded 16-row A matrix so all operand loads are branch-free.
// ---------------------------------------------------------------------------

#define H 4096
#define SDIM 256
#define NK 4
#define MPAD 16                // WMMA M dimension (candidate rows padded 4->16)
#define HCHUNKS 16
#define HCHUNK (H / HCHUNKS)   // 256
#define GT 256                 // block threads for streaming kernels
#define DPT 4                  // float4 per thread
#define DTILE (GT * DPT)       // 1024 columns per block
#define DTILES (H / DTILE)     // 4

#define ACT_SIGMOID 0
#define ACT_TANH    1
#define ACT_ID      2
#define ACT_MINUS   3

typedef __attribute__((ext_vector_type(2))) float v2f;
typedef __attribute__((ext_vector_type(8))) float v8f;

// --------------------------------------------------------------------------
// Batched GEMV partial: P[k][c][d] = sum_{h in chunk c} s[h] * M[k][h][d]
// Row-accumulate orientation -> perfectly coalesced b128 streams of M.
// --------------------------------------------------------------------------
__global__ void gemv_partial(const float* __restrict__ s,
                             const float* __restrict__ M,
                             float* __restrict__ P) {
  __shared__ float sh[HCHUNK];
  const int t  = threadIdx.x;
  const int dt = blockIdx.x;
  const int c  = blockIdx.y;
  const int k  = blockIdx.z;
  const int h0 = c * HCHUNK;

  for (int i = t; i < HCHUNK; i += GT) sh[i] = s[h0 + i];
  __syncthreads();

  const float* Mp = M + ((size_t)k * H + h0) * H + (size_t)dt * DTILE + t * DPT;
  float4 acc = make_float4(0.f, 0.f, 0.f, 0.f);

  for (int h = 0; h < HCHUNK; h += 4) {
    __builtin_prefetch(Mp + (size_t)16 * H, 0, 3);  // global_prefetch_b8
#pragma unroll
    for (int u = 0; u < 4; ++u) {
      const float4 row = *(const float4*)(Mp + (size_t)u * H);
      const float sv = sh[h + u];
      acc.x = fmaf(sv, row.x, acc.x);
      acc.y = fmaf(sv, row.y, acc.y);
      acc.z = fmaf(sv, row.z, acc.z);
      acc.w = fmaf(sv, row.w, acc.w);
    }
    Mp += (size_t)4 * H;
  }

  float* Pp = P + ((size_t)k * HCHUNKS + c) * H + (size_t)dt * DTILE + t * DPT;
  *(float4*)Pp = acc;
}

// --------------------------------------------------------------------------
// Zero the padded candidate rows 4..15 of VBUF once per call (rows 0..3 are
// rewritten by combine_act every round; rows 4..15 stay zero for WMMA).
// --------------------------------------------------------------------------
__global__ void init_pad(float* __restrict__ vbuf) {
  const int gid = blockIdx.x * blockDim.x + threadIdx.x;
  const size_t base = (size_t)NK * H + (size_t)gid * 4;  // rows 4..15
  *(float4*)(vbuf + base) = make_float4(0.f, 0.f, 0.f, 0.f);
}

// --------------------------------------------------------------------------
// Deterministic reduction over h-chunks: out[k][d] = sum_c P[k][c][d].
// Also zero-initializes the per-k maxabs slots (runs before every combine).
// --------------------------------------------------------------------------
__global__ void reduce_partials(const float* __restrict__ P,
                                float* __restrict__ out,
                                unsigned* __restrict__ maxabs) {
  const int gid = blockIdx.x * blockDim.x + threadIdx.x;  // NK*H/4 threads
  const size_t base = (size_t)gid * 4;
  const int k = (int)(base / H);
  const int d = (int)(base % H);
  float4 acc = make_float4(0.f, 0.f, 0.f, 0.f);
  for (int c = 0; c < HCHUNKS; ++c) {
    const float4 p = *(const float4*)(P + ((size_t)k * HCHUNKS + c) * H + d);
    acc.x += p.x; acc.y += p.y; acc.z += p.z; acc.w += p.w;
  }
  *(float4*)(out + base) = acc;
  if (gid < NK) maxabs[gid] = 0u;
}

// --------------------------------------------------------------------------
// res = (li op rj) + b ; v = act(res) ; maxabs[k] = max|res| (uint atomicMax
// on the bit pattern of |res| -> exact & deterministic).
// --------------------------------------------------------------------------
__device__ __forceinline__ float apply_act(float x, int act) {
  if (act == ACT_SIGMOID) return 1.f / (1.f + __expf(-x));
  if (act == ACT_TANH)    return tanhf(x);
  if (act == ACT_MINUS)   return 1.f - x;
  return x;
}

__global__ void combine_act(const float* __restrict__ li,
                            const float* __restrict__ rj,
                            const float* __restrict__ bias,
                            float* __restrict__ vbuf,
                            unsigned* __restrict__ maxabs,
                            int op, int act) {
  __shared__ float sred[GT];
  const int t = threadIdx.x;
  const int gid = blockIdx.x * blockDim.x + t;
  const size_t base = (size_t)gid * 4;
  const int k = (int)(base / H);   // each block lies within one k (1024 d's)

  float4 l = make_float4(0.f, 0.f, 0.f, 0.f);
  if (li) l = *(const float4*)(li + base);
  const float4 r  = *(const float4*)(rj + base);
  const float4 bb = *(const float4*)(bias + base);  // b is (NK,1,H) flat

  float4 res;
  if (op) { res.x = l.x * r.x; res.y = l.y * r.y; res.z = l.z * r.z; res.w = l.w * r.w; }
  else    { res.x = l.x + r.x; res.y = l.y + r.y; res.z = l.z + r.z; res.w = l.w + r.w; }
  res.x += bb.x; res.y += bb.y; res.z += bb.z; res.w += bb.w;

  float4 v;
  v.x = apply_act(res.x, act);
  v.y = apply_act(res.y, act);
  v.z = apply_act(res.z, act);
  v.w = apply_act(res.w, act);
  *(float4*)(vbuf + base) = v;

  float ma = fmaxf(fmaxf(fabsf(res.x), fabsf(res.y)), fmaxf(fabsf(res.z), fabsf(res.w)));
  sred[t] = ma;
  __syncthreads();
  for (int st = GT / 2; st > 0; st >>= 1) {
    if (t < st) sred[t] = fmaxf(sred[t], sred[t + st]);
    __syncthreads();
  }
  if (t == 0) atomicMax(&maxabs[k], __float_as_uint(sred[0]));
}

// --------------------------------------------------------------------------
// Scoring MLP + masked softmax + mixture, single workgroup (8 wave32s).
// scores[k] = W2 . relu(v_k @ W1 + b1) + b2 via v_wmma_f32_16x16x4_f32:
//   M=16 (padded candidate rows), N=16 (s-tile), K=4 (h).
// Each wave owns two adjacent s-tiles and shares one A load per h-step.
// A layout (16x4 f32): lanes 0-15 M=0..15 K={0,1}; lanes 16-31 M=0..15 K={2,3}.
// D layout (16x16 f32): VGPR j, lanes 0-15 -> row M=j, col N=lane.
// --------------------------------------------------------------------------
__global__ void score_mix(const float* __restrict__ vbuf,   // MPAD x H, rows 4..15 zero
                          const float* __restrict__ W1,
                          const float* __restrict__ b1,
                          const float* __restrict__ W2,
                          const float* __restrict__ b2,
                          unsigned* __restrict__ maxabs,
                          int maskmode,
                          float* __restrict__ gout) {
  __shared__ float psum[16 * 16 * NK];  // [s-tile][lane-n][m]
  __shared__ float sscore[NK];
  __shared__ float probs[NK];

  const int t = threadIdx.x;
  const int wave = t >> 5;
  const int lane = t & 31;
  const int m = lane & 15;
  const int koff = (lane < 16) ? 0 : 2;
  const int s0 = wave * 32;             // this wave's two s-tiles: s0, s0+16

  const float* aptr  = vbuf + (size_t)m * H + koff;
  const float* bptrA = W1 + (size_t)koff * SDIM + s0 + m;         // tile 0
  const float* bptrB = W1 + (size_t)koff * SDIM + s0 + 16 + m;    // tile 1

  v8f d0 = {0.f, 0.f, 0.f, 0.f, 0.f, 0.f, 0.f, 0.f};
  v8f d1 = {0.f, 0.f, 0.f, 0.f, 0.f, 0.f, 0.f, 0.f};

#pragma unroll 2
  for (int h0 = 0; h0 < H; h0 += 4) {
    const v2f a = *(const v2f*)(aptr + h0);           // single b64 load
    v2f bA, bB;
    bA.x = bptrA[(size_t)h0 * SDIM];
    bA.y = bptrA[(size_t)(h0 + 1) * SDIM];
    bB.x = bptrB[(size_t)h0 * SDIM];
    bB.y = bptrB[(size_t)(h0 + 1) * SDIM];
    d0 = __builtin_amdgcn_wmma_f32_16x16x4_f32(false, a, false, bA, (short)0,
                                               d0, false, false);
    d1 = __builtin_amdgcn_wmma_f32_16x16x4_f32(false, a, false, bB, (short)0,
                                               d1, false, false);
  }

  float darr0[8], darr1[8];
  *(v8f*)darr0 = d0;
  *(v8f*)darr1 = d1;
  if (lane < 16) {
    {
      const int n = s0 + lane;
      const float w2 = W2[n];
      const float bb = b1[n];
#pragma unroll
      for (int mm = 0; mm < NK; ++mm) {
        const float hv = fmaxf(darr0[mm] + bb, 0.f);
        psum[((wave * 2) * 16 + lane) * NK + mm] = hv * w2;
      }
    }
    {
      const int n = s0 + 16 + lane;
      const float w2 = W2[n];
      const float bb = b1[n];
#pragma unroll
      for (int mm = 0; mm < NK; ++mm) {
        const float hv = fmaxf(darr1[mm] + bb, 0.f);
        psum[((wave * 2 + 1) * 16 + lane) * NK + mm] = hv * w2;
      }
    }
  }
  __syncthreads();

  if (t < NK) {  // fixed-order sum over all 256 hidden units -> deterministic
    float sc = b2[0];
    for (int j = 0; j < 256; ++j) sc += psum[j * NK + t];
    const bool mask = (maskmode == 0) || (t == 3) ||
                      (__uint_as_float(maxabs[t]) < 1.0f);
    sscore[t] = mask ? sc : -__builtin_inff();
    maxabs[t] = 0u;  // reset for the next round's combine
  }
  __syncthreads();

  if (t == 0) {
    float mx = -__builtin_inff();
    for (int k2 = 0; k2 < NK; ++k2) mx = fmaxf(mx, sscore[k2]);
    float e[NK]; float Z = 0.f;
    for (int k2 = 0; k2 < NK; ++k2) { e[k2] = __expf(sscore[k2] - mx); Z += e[k2]; }
    for (int k2 = 0; k2 < NK; ++k2) probs[k2] = e[k2] / Z;
  }
  __syncthreads();

  const float p0 = probs[0], p1 = probs[1], p2 = probs[2], p3 = probs[3];
  for (int d0i = t * 4; d0i < H; d0i += 256 * 4) {
    const float4 v0 = *(const float4*)(vbuf + d0i);
    const float4 v1 = *(const float4*)(vbuf + H + d0i);
    const float4 v2 = *(const float4*)(vbuf + 2 * H + d0i);
    const float4 v3 = *(const float4*)(vbuf + 3 * H + d0i);
    float4 g;
    g.x = p0 * v0.x + p1 * v1.x + p2 * v2.x + p3 * v3.x;
    g.y = p0 * v0.y + p1 * v1.y + p2 * v2.y + p3 * v3.y;
    g.z = p0 * v0.z + p1 * v1.z + p2 * v2.z + p3 * v3.z;
    g.w = p0 * v0.w + p1 * v1.w + p2 * v2.w + p3 * v3.w;
    *(float4*)(gout + d0i) = g;
  }
}

// --------------------------------------------------------------------------
extern "C" void kernel_launch(void* const* d_in, const int* in_sizes, int n_in,
                              void* d_out, int out_size, void* d_ws, size_t ws_size,
                              hipStream_t stream) {
  const float* x     = (const float*)d_in[0];
  const float* hprev = (const float*)d_in[1];
  const float* L     = (const float*)d_in[2];
  const float* R     = (const float*)d_in[3];
  const float* bias  = (const float*)d_in[4];
  const float* W1    = (const float*)d_in[5];
  const float* b1    = (const float*)d_in[6];
  const float* W2    = (const float*)d_in[7];
  const float* W2b   = (const float*)d_in[8];
  float* out = (float*)d_out;   // (9,1,H)
  float* ws  = (float*)d_ws;

  // workspace layout (floats): ~1.7 MB total
  const size_t NPART = (size_t)NK * HCHUNKS * H;  // 262144
  const size_t PROD  = (size_t)NK * H;            // 16384
  float* P     = ws;
  float* LIX   = P + NPART;       // cached x@L   (r0, r4)
  float* LICUR = LIX + PROD;      // h@L / G5@L / G6@L
  float* RJG0  = LICUR + PROD;    // cached G1@R == G0@R (r3, r5)
  float* RJG3  = RJG0 + PROD;     // cached G3@R (r4, r8 since G7==G3)
  float* RJCUR = RJG3 + PROD;     // h@R / G4@R
  float* VBUF  = RJCUR + PROD;    // MPAD x H candidates (rows 4..15 zero)
  unsigned* MAXABS = (unsigned*)(VBUF + (size_t)MPAD * H);

  const dim3 ggrid(DTILES, HCHUNKS, NK);
  const int rblocks = (NK * H / 4) / GT;           // 16
  const int padblocks = ((MPAD - NK) * H / 4) / GT; // 48

  auto GEMV = [&](const float* s, const float* M, float* prod) {
    gemv_partial<<<ggrid, GT, 0, stream>>>(s, M, P);
    reduce_partials<<<rblocks, GT, 0, stream>>>(P, prod, MAXABS);
  };
  auto COMBINE = [&](const float* li, const float* rj, int op, int act) {
    combine_act<<<rblocks, GT, 0, stream>>>(li, rj, bias, VBUF, MAXABS, op, act);
  };
  auto SCORE = [&](int maskmode, int row) {
    score_mix<<<1, 256, 0, stream>>>(VBUF, W1, b1, W2, W2b, MAXABS, maskmode,
                                     out + (size_t)row * H);
  };
  const size_t ROWB = (size_t)H * sizeof(float);

  init_pad<<<padblocks, GT, 0, stream>>>(VBUF);

  // r0: x@L + h@R, sigmoid.  (r1, r2 are bitwise-identical -> copies)
  GEMV(x, L, LIX);
  GEMV(hprev, R, RJCUR);
  COMBINE(LIX, RJCUR, 0, ACT_SIGMOID);
  SCORE(0, 0);
  hipMemcpyAsync(out + 1 * H, out, ROWB, hipMemcpyDeviceToDevice, stream);  // G1
  hipMemcpyAsync(out + 2 * H, out, ROWB, hipMemcpyDeviceToDevice, stream);  // G2

  // r3: h@L * G1@R, identity(masked)
  GEMV(hprev, L, LICUR);
  GEMV(out + 1 * H, R, RJG0);
  COMBINE(LICUR, RJG0, 1, ACT_ID);
  SCORE(1, 3);

  // r4: x@L + G3@R, tanh   (x@L cached)
  GEMV(out + 3 * H, R, RJG3);
  COMBINE(LIX, RJG3, 0, ACT_TANH);
  SCORE(0, 4);

  // r5: 0 + G0@R, minus    (G0@R == G1@R cached; L-side GEMV skipped)
  COMBINE(nullptr, RJG0, 0, ACT_MINUS);
  SCORE(1, 5);

  // r6: G5@L * G4@R, identity(masked)
  GEMV(out + 5 * H, L, LICUR);
  GEMV(out + 4 * H, R, RJCUR);
  COMBINE(LICUR, RJCUR, 1, ACT_ID);
  SCORE(1, 6);

  // r7: h@L * G2@R == r3 bitwise (G2==G1) -> copy
  hipMemcpyAsync(out + 7 * H, out + 3 * H, ROWB, hipMemcpyDeviceToDevice, stream);

  // r8: G6@L + G7@R, identity(masked)   (G7@R == G3@R cached)
  GEMV(out + 6 * H, L, LICUR);
  COMBINE(LICUR, RJG3, 0, ACT_ID);
  SCORE(1, 8);
}